// GnBlock_45509473468813
// MI455X (gfx1250) — compile-verified
//
#include <hip/hip_runtime.h>
#include <hip/hip_bf16.h>

// ---------------------------------------------------------------------------
// ChebConv (K=5, sym norm, lambda_max=2) for MI455X / gfx1250.
//   Phase 1: degree scatter, rsqrt, per-edge norm           (L2-resident)
//   Phase 2: 4x propagation via wave-per-edge gather + f32 atomic scatter,
//            Chebyshev recurrence folded into buffer pre-init (-Tx_{k-2})
//   Phase 3: out = b + sum_k Tx_k @ W[k] using V_WMMA_F32_16X16X4_F32,
//            W staged in LDS, one wave per 16x64 output tile.
// ---------------------------------------------------------------------------

typedef __attribute__((ext_vector_type(2))) float v2f;
typedef __attribute__((ext_vector_type(8))) float v8f;

static constexpr int DF = 64;   // feature dim

// ---------------------------------------------------------------- utilities
__global__ void fill_kernel(float* __restrict__ p, float v, int n) {
  int i = blockIdx.x * blockDim.x + threadIdx.x;
  if (i < n) p[i] = v;
}

// dst = -src   (in-place safe: each thread touches only its own element)
__global__ void negcopy_kernel(float* __restrict__ dst, const float* __restrict__ src, int n) {
  int i = blockIdx.x * blockDim.x + threadIdx.x;
  if (i < n) dst[i] = -src[i];
}

__global__ void init_out_kernel(float* __restrict__ out, const float* __restrict__ b, int n) {
  int i = blockIdx.x * blockDim.x + threadIdx.x;
  if (i < n) out[i] = b[i & (DF - 1)];
}

// ------------------------------------------------------------ degree / norm
__global__ void deg_kernel(float* __restrict__ deg, const int* __restrict__ src, int E) {
  int e = blockIdx.x * blockDim.x + threadIdx.x;
  if (e < E) atomicAdd(&deg[src[e]], 1.0f);
}

__global__ void rsqrt_kernel(float* __restrict__ dis, int n) {
  int i = blockIdx.x * blockDim.x + threadIdx.x;
  if (i < n) {
    float g = dis[i];
    dis[i] = (g > 0.0f) ? rsqrtf(g) : 0.0f;
  }
}

__global__ void norm_kernel(float* __restrict__ norm, const float* __restrict__ dis,
                            const int* __restrict__ src, const int* __restrict__ dst, int E) {
  int e = blockIdx.x * blockDim.x + threadIdx.x;
  if (e < E) norm[e] = -dis[src[e]] * dis[dst[e]];
}

// -------------------------------------------------------------- propagation
// One wave per edge. Lanes cover 64 features as float2 (256B coalesced gather,
// 2 x global_atomic_add_f32 scatter per lane). Buffers are L2-resident (25.6MB).
__global__ void __launch_bounds__(256)
prop_kernel(const float* __restrict__ h, float* __restrict__ acc,
            const int* __restrict__ src, const int* __restrict__ dst,
            const float* __restrict__ norm, float coef, int E) {
  int gid  = blockIdx.x * blockDim.x + threadIdx.x;
  int edge = gid >> 5;
  int lane = gid & 31;
  if (edge >= E) return;
  int s = src[edge];
  int d = dst[edge];
  float c = coef * norm[edge];
  float2 v = *(const float2*)(h + (size_t)s * DF + lane * 2);
  float* op = acc + (size_t)d * DF + lane * 2;
  atomicAdd(op + 0, c * v.x);
  atomicAdd(op + 1, c * v.y);
}

// ------------------------------------------------------------ WMMA GEMM-acc
// out[16x64 tile] += A[16x64 tile] @ W[64x64]     (fp32, 16x16x4 WMMA)
//
// C/D layout (v8f): lane L, vgpr v -> M = v + (L>=16 ? 8:0), N = L%16
// A   layout (v2f): lane L -> M = L%16, holds K = k0 + (L>=16 ? 2:0) + {0,1}
// B   layout (v2f): lane L -> N = L%16, holds K = k0 + (L>=16 ? 2:0) + {0,1}
__global__ void __launch_bounds__(256)
gemm_acc_kernel(const float* __restrict__ A, const float* __restrict__ W,
                float* __restrict__ out, int ntiles) {
  __shared__ float Wl[DF * DF];            // 16 KB of the 320 KB WGP LDS
  {
    const float4* s = (const float4*)W;
    float4* d = (float4*)Wl;
    for (int i = threadIdx.x; i < DF * DF / 4; i += blockDim.x) d[i] = s[i];
  }
  __syncthreads();

  int lane = threadIdx.x & 31;
  int tile = blockIdx.x * (blockDim.x >> 5) + (threadIdx.x >> 5);
  if (tile >= ntiles) return;             // wave-uniform: EXEC stays all-ones

  int l16  = lane & 15;
  int half = (lane >> 4) & 1;
  size_t row0 = (size_t)tile * 16;

  // load current accumulator tile (read-modify-write so GEMMs chain over k)
  v8f c[4];
#pragma unroll
  for (int t = 0; t < 4; ++t) {
    int n = t * 16 + l16;
#pragma unroll
    for (int v = 0; v < 8; ++v)
      c[t][v] = out[(row0 + v + half * 8) * DF + n];
  }

  const float* arow = A + (row0 + l16) * DF + half * 2;
#pragma unroll
  for (int k0 = 0; k0 < DF; k0 += 4) {
    float2 av = *(const float2*)(arow + k0);
    v2f a; a[0] = av.x; a[1] = av.y;
    int krow = k0 + half * 2;
#pragma unroll
    for (int t = 0; t < 4; ++t) {
      int n = t * 16 + l16;
      v2f b;
      b[0] = Wl[krow * DF + n];
      b[1] = Wl[(krow + 1) * DF + n];
      c[t] = __builtin_amdgcn_wmma_f32_16x16x4_f32(
          /*neg_a=*/false, a, /*neg_b=*/false, b,
          /*c_mod=*/(short)0, c[t], /*reuse_a=*/false, /*reuse_b=*/false);
    }
  }

#pragma unroll
  for (int t = 0; t < 4; ++t) {
    int n = t * 16 + l16;
#pragma unroll
    for (int v = 0; v < 8; ++v)
      out[(row0 + v + half * 8) * DF + n] = c[t][v];
  }
}

// ------------------------------------------------------------------- launch
extern "C" void kernel_launch(void* const* d_in, const int* in_sizes, int n_in,
                              void* d_out, int out_size, void* d_ws, size_t ws_size,
                              hipStream_t stream) {
  const float* x    = (const float*)d_in[0];   // [N, 64]
  const int*   eidx = (const int*)d_in[1];     // [2, E]  row0=src, row1=dst
  const float* W    = (const float*)d_in[2];   // [K, 64, 64]
  const float* b    = (const float*)d_in[3];   // [64]
  float* out        = (float*)d_out;           // [N, 64]

  const int N  = in_sizes[0] / DF;
  const int E  = in_sizes[1] / 2;
  const int K  = in_sizes[2] / (DF * DF);
  const int ND = N * DF;

  const int* src = eidx;
  const int* dst = eidx + E;

  // workspace layout (floats): dis[N] | norm[E] | bufA[N*64] | bufB[N*64]
  float* dis  = (float*)d_ws;
  float* norm = dis + N;
  float* bufA = norm + E;
  float* bufB = bufA + ND;

  const int BS = 256;
  auto blk = [&](long n) { return (unsigned)((n + BS - 1) / BS); };
  const int ntiles     = (N + 15) / 16;
  const int gemm_grid  = (ntiles + (BS / 32) - 1) / (BS / 32);
  const unsigned prop_grid = blk((long)E * 32);

  // ---- degree -> dis -> per-edge norm
  fill_kernel<<<blk(N), BS, 0, stream>>>(dis, 0.0f, N);
  deg_kernel<<<blk(E), BS, 0, stream>>>(dis, src, E);
  rsqrt_kernel<<<blk(N), BS, 0, stream>>>(dis, N);
  norm_kernel<<<blk(E), BS, 0, stream>>>(norm, dis, src, dst, E);

  // ---- out = b + x @ W[0]
  init_out_kernel<<<blk(ND), BS, 0, stream>>>(out, b, ND);
  gemm_acc_kernel<<<gemm_grid, BS, 0, stream>>>(x, W, out, ntiles);

  // ---- Tx1 = prop(x);  out += Tx1 @ W[1]
  fill_kernel<<<blk(ND), BS, 0, stream>>>(bufA, 0.0f, ND);
  prop_kernel<<<prop_grid, BS, 0, stream>>>(x, bufA, src, dst, norm, 1.0f, E);
  if (K > 1)
    gemm_acc_kernel<<<gemm_grid, BS, 0, stream>>>(bufA, W + (size_t)1 * DF * DF, out, ntiles);

  // ---- Tx_k = 2*prop(Tx_{k-1}) - Tx_{k-2};  out += Tx_k @ W[k]
  const float* Tpp = x;      // Tx_{k-2}
  float*       Tp  = bufA;   // Tx_{k-1}
  float*       Tn  = bufB;   // target (aliases Tx_{k-2}'s buffer for k>=3)
  for (int k = 2; k < K; ++k) {
    negcopy_kernel<<<blk(ND), BS, 0, stream>>>(Tn, Tpp, ND);   // Tn = -Tx_{k-2}
    prop_kernel<<<prop_grid, BS, 0, stream>>>(Tp, Tn, src, dst, norm, 2.0f, E);
    gemm_acc_kernel<<<gemm_grid, BS, 0, stream>>>(Tn, W + (size_t)k * DF * DF, out, ntiles);
    Tpp = Tp;
    float* t = Tp; Tp = Tn; Tn = t;
  }
}